// DeformableConvNet_46789373723354
// MI455X (gfx1250) — compile-verified
//
#include <hip/hip_runtime.h>

// ---------------------------------------------------------------------------
// Deformable 3D conv net on gfx1250 (MI455X), fp32 WMMA path.
//   B=2, C=32, O=32, S=48, taps=27, offset channels=81.
// All GEMM work lowered to V_WMMA_F32_16X16X4_F32 (exact fp32).
// K ordering for the offset conv is tap-major (k = tap*32 + c) so a 4-wide
// WMMA K-chunk never straddles a tap. Zero-padding is done branch-free:
// clamp the source offset to a legal address once per tap and multiply the
// loaded value by a 0/1 mask, so all 16 chunk loads stay in one s_clause.
// ---------------------------------------------------------------------------

typedef __attribute__((ext_vector_type(2))) float v2f;   // A/B fragment: 16x4 / 4x16 f32
typedef __attribute__((ext_vector_type(8))) float v8f;   // C/D fragment: 16x16 f32

#define SDIM 48
#define SS   (SDIM * SDIM)
#define NVOX (SDIM * SDIM * SDIM)      // 110592
#define NB   2
#define CIN  32
#define OCH  32
#define KV   27
#define OFFC 81                        // 3 * KV
#define KTOT (CIN * KV)                // 864
#define MPAD 96                        // 81 padded to 6 WMMA M-tiles
#define TILES_PER_B (NVOX / 16)        // 6912
#define NTILES (NB * TILES_PER_B)      // 13824

// workspace layout (floats)
#define WS_OFFBUF 0                                   // B*81*NVOX offsets
#define WS_WOFF   (NB * OFFC * NVOX)                  // packed w_off [k=tap*32+c][m], m<96
#define WS_WDC    (WS_WOFF + KTOT * MPAD)             // packed w_dc  [tap][c][o]

// ---------------------------------------------------------------------------
// Kernel 0: repack weights so A-fragment loads are lane-coalesced and the
// K ordering is tap-major.
//   wOffP[(tap*32+c)*96 + m] = (m < 81) ? w_off[m*864 + c*27 + tap] : 0
//   wDcP[(tap*32 + c)*32 + o] = w_dc[o*864 + c*27 + tap]
// ---------------------------------------------------------------------------
__global__ void repack_weights(const float* __restrict__ w_off,
                               const float* __restrict__ w_dc,
                               float* __restrict__ ws) {
  int i = blockIdx.x * blockDim.x + threadIdx.x;
  if (i < KTOT * MPAD) {
    int kk = i / MPAD, m = i % MPAD;
    int tap = kk >> 5, c = kk & 31;
    ws[WS_WOFF + i] = (m < OFFC) ? w_off[m * KTOT + c * KV + tap] : 0.0f;
  }
  int j = i - KTOT * MPAD;
  if (j >= 0 && j < KV * CIN * OCH) {
    int o = j % OCH;
    int c = (j / OCH) % CIN;
    int tap = j / (OCH * CIN);
    ws[WS_WDC + j] = w_dc[o * KTOT + c * KV + tap];
  }
}

// ---------------------------------------------------------------------------
// Kernel 1: offset conv as im2col GEMM, M=96(81) x K=864 x N=16 per wave.
// Outer loop over 27 taps (spatial decode + pad mask hoisted, mostly SALU);
// inner 8 K-chunks are unconditional NVOX-strided loads (offset clamped to a
// legal address) * mask, feeding 6 M-tiles of V_WMMA_F32_16X16X4_F32.
// ---------------------------------------------------------------------------
__global__ void offset_conv(const float* __restrict__ x,
                            const float* __restrict__ b_off,
                            float* __restrict__ ws) {
  const float* wOffP = ws + WS_WOFF;
  float* offBuf = ws + WS_OFFBUF;

  const int lane = threadIdx.x & 31;
  const int wave = threadIdx.x >> 5;
  const int tile = blockIdx.x * (blockDim.x >> 5) + wave;
  const int b  = tile / TILES_PER_B;
  const int p0 = (tile % TILES_PER_B) * 16;
  const int l  = lane & 15;        // N index (and M index for A-frag)
  const int hf = lane >> 4;        // half: selects K pair within chunk
  const int d  = p0 / SS;
  const int h  = (p0 / SDIM) % SDIM;
  const int w0 = p0 % SDIM;        // tile never crosses a W row (48 % 16 == 0)
  const int wv = w0 + l;
  const float* xb = x + (size_t)b * CIN * NVOX;

  v8f zero = {};
  v8f acc[6];
#pragma unroll
  for (int mt = 0; mt < 6; ++mt) acc[mt] = zero;

  for (int tap = 0; tap < KV; ++tap) {
    // spatial decode: uniform per wave except the per-lane W coordinate
    const int kd = tap / 9 - 1, kh = (tap / 3) % 3 - 1, kw = tap % 3 - 1;
    const int dd = d + kd, hh = h + kh, wp = wv + kw;
    const bool ok = ((unsigned)dd < SDIM) && ((unsigned)hh < SDIM) &&
                    ((unsigned)wp < SDIM);
    // Branch-free zero padding: always load from a legal address, then mask.
    const int   sofs = ok ? (dd * SDIM + hh) * SDIM + wp : 0;
    const float msk  = ok ? 1.0f : 0.0f;
    const int kbase = tap * CIN;

#pragma unroll
    for (int cc = 0; cc < 8; ++cc) {
      const int c0 = cc * 4 + 2 * hf;                // channel pair for this lane
      float v0 = xb[(c0    ) * NVOX + sofs];
      float v1 = xb[(c0 + 1) * NVOX + sofs];
      v2f bvec;
      bvec[0] = msk * v0;
      bvec[1] = msk * v1;
      const int kk0 = kbase + c0;
#pragma unroll
      for (int mt = 0; mt < 6; ++mt) {
        v2f avec;
        avec[0] = wOffP[(kk0    ) * MPAD + mt * 16 + l];
        avec[1] = wOffP[(kk0 + 1) * MPAD + mt * 16 + l];
        acc[mt] = __builtin_amdgcn_wmma_f32_16x16x4_f32(
            false, avec, false, bvec, (short)0, acc[mt], false, false);
      }
    }
  }

  // Epilogue: D layout — VGPR v holds row (v + 8*hf) of this M-tile.
  // M-tiles 0..4 cover rows 0..79: all valid, store unconditionally.
#pragma unroll
  for (int mt = 0; mt < 5; ++mt) {
#pragma unroll
    for (int v = 0; v < 8; ++v) {
      int row = mt * 16 + v + 8 * hf;
      offBuf[((size_t)b * OFFC + row) * NVOX + p0 + l] = acc[mt][v] + b_off[row];
    }
  }
  // M-tile 5 covers rows 80..95: only row 80 (v==0, hf==0) is real.
  if (hf == 0)
    offBuf[((size_t)b * OFFC + 80) * NVOX + p0 + l] = acc[5][0] + b_off[80];
}

// ---------------------------------------------------------------------------
// Kernel 2: deformable stage. Per tap: trilinear 8-corner gather feeds the
// B fragment; w_dc[:, :, tap] (32x32) feeds A; accumulate 27 taps via WMMA.
// Gather is already branch-free (clamped indices, weight zeroed outside).
// ---------------------------------------------------------------------------
__global__ void deform_conv(const float* __restrict__ x,
                            const float* __restrict__ b_dc,
                            const float* __restrict__ ws,
                            float* __restrict__ out) {
  const float* wDcP   = ws + WS_WDC;
  const float* offBuf = ws + WS_OFFBUF;

  const int lane = threadIdx.x & 31;
  const int wave = threadIdx.x >> 5;
  const int tile = blockIdx.x * (blockDim.x >> 5) + wave;
  const int b  = tile / TILES_PER_B;
  const int p0 = (tile % TILES_PER_B) * 16;
  const int l  = lane & 15;
  const int hf = lane >> 4;
  const int d  = p0 / SS;
  const int h  = (p0 / SDIM) % SDIM;
  const int w0 = p0 % SDIM;
  const float* xb = x + (size_t)b * CIN * NVOX;

  v8f acc0 = {}, acc1 = {};

  for (int tap = 0; tap < KV; ++tap) {
    const int kd = tap / 9 - 1, kh = (tap / 3) % 3 - 1, kw = tap % 3 - 1;
    const int p = p0 + l;
    // learned offsets (channel = tap*3 + axis), coalesced per half-wave
    float od = offBuf[((size_t)b * OFFC + tap * 3 + 0) * NVOX + p];
    float oh = offBuf[((size_t)b * OFFC + tap * 3 + 1) * NVOX + p];
    float ow = offBuf[((size_t)b * OFFC + tap * 3 + 2) * NVOX + p];
    float pd = (float)(d + kd) + od;
    float ph = (float)(h + kh) + oh;
    float pw = (float)(w0 + l + kw) + ow;
    float fd = floorf(pd), fh = floorf(ph), fw = floorf(pw);

    // 8 trilinear corners: weight (0 outside volume) + clamped flat index.
    float wcor[8];
    int   icor[8];
#pragma unroll
    for (int i = 0; i < 8; ++i) {
      float di = fd + (float)((i >> 2) & 1);
      float hi = fh + (float)((i >> 1) & 1);
      float wi = fw + (float)(i & 1);
      float wgt = (1.0f - fabsf(pd - di)) * (1.0f - fabsf(ph - hi)) *
                  (1.0f - fabsf(pw - wi));
      bool okc = (di >= 0.0f) && (di < 48.0f) && (hi >= 0.0f) && (hi < 48.0f) &&
                 (wi >= 0.0f) && (wi < 48.0f);
      float dc = fminf(fmaxf(di, 0.0f), 47.0f);
      float hc = fminf(fmaxf(hi, 0.0f), 47.0f);
      float wc = fminf(fmaxf(wi, 0.0f), 47.0f);
      icor[i] = ((int)dc * SDIM + (int)hc) * SDIM + (int)wc;
      wcor[i] = okc ? wgt : 0.0f;
    }

    // K = 32 channels -> 8 chunks of 4; two M-tiles (O = 32).
#pragma unroll
    for (int cc = 0; cc < 8; ++cc) {
      const int c0 = cc * 4 + 2 * hf;
      v2f bvec;
#pragma unroll
      for (int j = 0; j < 2; ++j) {
        const float* xc = xb + (c0 + j) * NVOX;
        float v = 0.0f;
#pragma unroll
        for (int i = 0; i < 8; ++i) v += wcor[i] * xc[icor[i]];
        bvec[j] = v;
      }
      v2f a0, a1;
      a0[0] = wDcP[(tap * CIN + c0    ) * OCH + l];
      a0[1] = wDcP[(tap * CIN + c0 + 1) * OCH + l];
      a1[0] = wDcP[(tap * CIN + c0    ) * OCH + 16 + l];
      a1[1] = wDcP[(tap * CIN + c0 + 1) * OCH + 16 + l];
      acc0 = __builtin_amdgcn_wmma_f32_16x16x4_f32(
          false, a0, false, bvec, (short)0, acc0, false, false);
      acc1 = __builtin_amdgcn_wmma_f32_16x16x4_f32(
          false, a1, false, bvec, (short)0, acc1, false, false);
    }
  }

  // Epilogue: add bias, store coalesced per half-wave.
#pragma unroll
  for (int v = 0; v < 8; ++v) {
    int o0 = v + 8 * hf;
    out[((size_t)b * OCH + o0) * NVOX + p0 + l] = acc0[v] + b_dc[o0];
    int o1 = 16 + v + 8 * hf;
    out[((size_t)b * OCH + o1) * NVOX + p0 + l] = acc1[v] + b_dc[o1];
  }
}

// ---------------------------------------------------------------------------
extern "C" void kernel_launch(void* const* d_in, const int* in_sizes, int n_in,
                              void* d_out, int out_size, void* d_ws, size_t ws_size,
                              hipStream_t stream) {
  (void)in_sizes; (void)n_in; (void)out_size; (void)ws_size;
  const float* x     = (const float*)d_in[0];
  const float* w_off = (const float*)d_in[1];
  const float* b_off = (const float*)d_in[2];
  const float* w_dc  = (const float*)d_in[3];
  const float* b_dc  = (const float*)d_in[4];
  float* out = (float*)d_out;
  float* ws  = (float*)d_ws;

  // Stage 0: weight repack (coalesced A-fragments, tap-major K, M padded to 96).
  {
    int total = KTOT * MPAD + KV * CIN * OCH;   // 82944 + 27648
    int blocks = (total + 255) / 256;
    repack_weights<<<blocks, 256, 0, stream>>>(w_off, w_dc, ws);
  }
  // Stage 1: offset conv (im2col GEMM via fp32 WMMA) -> workspace.
  offset_conv<<<NTILES / 4, 128, 0, stream>>>(x, b_off, ws);
  // Stage 2: deformable gather + tap-GEMM accumulation via fp32 WMMA.
  deform_conv<<<NTILES / 4, 128, 0, stream>>>(x, b_dc, ws, out);
}